// QuantumLayer_59751585022689
// MI455X (gfx1250) — compile-verified
//
#include <hip/hip_runtime.h>
#include <hip/hip_bf16.h>

// ---------------------------------------------------------------------------
// 4-qubit variational circuit, B = 2^20 samples.
//
// out[b] = r(b)^T * A * r(b)
//   r(b)[n] = prod_i ( bit_i(n) ? sin(x[b,i]/2) : cos(x[b,i]/2) ),  qubit0 = MSB
//   A       = fixed real 16x16 matrix folding RZ layer + CNOT chain + RX layer
//             + <Z0> measurement + the (-i)^popcount encoding phases.
//
// Roofline: 16B in + 4B out per sample -> ~20MB @ 23.3TB/s ~ 0.9us floor.
//
// Transposed mapping: Y = A * R^T on V_WMMA_F32_16X16X4_F32 (samples on the
// N dimension). Lane (h,l) then holds Y[v+8h][sample l] in acc[v], so the
// quadratic form finishes with 8 in-lane FMAs + ONE cross-half shuffle.
// No LDS tile, no barriers, no butterfly reduction.
// ---------------------------------------------------------------------------

typedef float v2f __attribute__((ext_vector_type(2)));
typedef float v8f __attribute__((ext_vector_type(8)));

#define TILES_PER_WAVE 4
#define WAVES_PER_BLOCK 8

struct cplx { float re, im; };
__device__ __forceinline__ cplx cmul(cplx a, cplx b) {
    return { a.re * b.re - a.im * b.im, a.re * b.im + a.im * b.re };
}
__device__ __forceinline__ cplx cconj(cplx a) { return { a.re, -a.im }; }

// Column entry col_k[m] of the fixed sub-circuit U = RXw * P_cnot * D_rz.
__device__ __forceinline__ cplx col_entry(int k, int m,
                                          const float* rzw,
                                          const float* cw, const float* sw)
{
    // RZ diagonal phase on original basis index k (qubit0 = MSB)
    int b0 = (k >> 3) & 1, b1 = (k >> 2) & 1, b2 = (k >> 1) & 1, b3 = k & 1;
    float phi = 0.5f * ((b0 ? rzw[0] : -rzw[0]) + (b1 ? rzw[1] : -rzw[1]) +
                        (b2 ? rzw[2] : -rzw[2]) + (b3 ? rzw[3] : -rzw[3]));
    // CNOT chain permutation: (0,1),(1,2),(2,3): target ^= control
    b1 ^= b0; b2 ^= b1; b3 ^= b2;
    int kp = (b0 << 3) | (b1 << 2) | (b2 << 1) | b3;

    // RX layer (separable): entry = mag * (-i)^d, d = popcount(m ^ kp)
    int xr = m ^ kp;
    int d  = __popc(xr) & 3;
    float mag = (((xr >> 3) & 1) ? sw[0] : cw[0]) *
                (((xr >> 2) & 1) ? sw[1] : cw[1]) *
                (((xr >> 1) & 1) ? sw[2] : cw[2]) *
                (( xr       & 1) ? sw[3] : cw[3]);

    float cp, sp; sincosf(phi, &sp, &cp);
    cplx ph = { cp, sp };                                  // e^{i phi}
    static const float miR[4] = { 1.f,  0.f, -1.f, 0.f };  // (-i)^d
    static const float miI[4] = { 0.f, -1.f,  0.f, 1.f };
    cplx mi = { miR[d], miI[d] };
    cplx v = cmul(ph, mi);
    return { mag * v.re, mag * v.im };
}

// One block, 256 threads: thread (j,k) -> A[j*16+k]. Runs once per launch.
__global__ void qc_setup(const float* __restrict__ w, float* __restrict__ A)
{
    int t = threadIdx.x;
    int j = t >> 4, k = t & 15;

    float rzw[4], cw[4], sw[4];
#pragma unroll
    for (int i = 0; i < 4; ++i) {
        rzw[i] = w[i];
        sincosf(0.5f * w[4 + i], &sw[i], &cw[i]);
    }

    // M_jk = sum_m z_m * conj(col_j[m]) * col_k[m],  z_m = (m<8 ? +1 : -1)
    cplx M = { 0.f, 0.f };
#pragma unroll
    for (int m = 0; m < 16; ++m) {
        cplx cj = col_entry(j, m, rzw, cw, sw);
        cplx ck = col_entry(k, m, rzw, cw, sw);
        cplx p  = cmul(cconj(cj), ck);
        float z = (m < 8) ? 1.f : -1.f;
        M.re += z * p.re;
        M.im += z * p.im;
    }

    // A_jk = Re( i^{pop(j)} * (-i)^{pop(k)} * M ) = Re( i^e * M )
    int e = (__popc(j) - __popc(k)) & 3;
    float val = (e == 0) ?  M.re :
                (e == 1) ? -M.im :
                (e == 2) ? -M.re :  M.im;
    A[j * 16 + k] = val;
}

// Main kernel: 8 waves/block, each wave owns TILES_PER_WAVE 16-sample tiles.
__global__ void __launch_bounds__(256)
qc_main(const float* __restrict__ x, const float* __restrict__ Amat,
        float* __restrict__ out, int B)
{
    const int lane  = threadIdx.x & 31;
    const int wv    = threadIdx.x >> 5;
    const int gw    = blockIdx.x * WAVES_PER_BLOCK + wv;
    const int tile0 = gw * TILES_PER_WAVE;
    const int h     = lane >> 4;                   // half-wave 0|1
    const int l     = lane & 15;

    // A-operand = fixed matrix A, loaded once per wave, reused for all tiles.
    // Layout: lane -> row M = l; VGPR v -> K_local = 2h + v.
    // Adjacent cols -> one 8B-aligned float2 load per chunk.
    v2f aop[4];
#pragma unroll
    for (int q = 0; q < 4; ++q) {
        float2 t2 = *reinterpret_cast<const float2*>(
            &Amat[l * 16 + 4 * q + 2 * h]);
        aop[q].x = t2.x;
        aop[q].y = t2.y;
    }

    for (int it = 0; it < TILES_PER_WAVE; ++it) {
        const int base = (tile0 + it) * 16;
        int m = base + l; if (m >= B) m = B - 1;   // clamp tail; store guarded

        float4 xv = reinterpret_cast<const float4*>(x)[m];
        float c0, s0, c1, s1, c2, s2, c3, s3;
        __sincosf(0.5f * xv.x, &s0, &c0);
        __sincosf(0.5f * xv.y, &s1, &c1);
        __sincosf(0.5f * xv.z, &s2, &c2);
        __sincosf(0.5f * xv.w, &s3, &c3);

        // --- Y = A * R^T via 4x V_WMMA_F32_16X16X4_F32 (K = 16) ---
        // B-operand element (K = 4q+2h+j, N = l) = r_l[4q+2h+j]:
        //   bit3 = q&2, bit2 = q&1 (compile-time), bit1 = h (1 cndmask), bit0 = j
        float g2h = h ? s2 : c2;
        v8f acc = {};
#pragma unroll
        for (int q = 0; q < 4; ++q) {
            float g01 = ((q & 2) ? s0 : c0) * ((q & 1) ? s1 : c1); // static sel
            float b3  = g01 * g2h;
            v2f b;
            b.x = b3 * c3;    // j = 0
            b.y = b3 * s3;    // j = 1
            acc = __builtin_amdgcn_wmma_f32_16x16x4_f32(
                /*neg_a=*/false, aop[q], /*neg_b=*/false, b,
                /*c_mod=*/(short)0, acc, /*reuse_a=*/false, /*reuse_b=*/false);
        }

        // --- out[l] = sum_j r_l[j] * Y[j][l] ---
        // D layout: acc[v] at lane (h,l) = Y[j = v+8h][sample l].
        // r_l[v+8h] = (h ? s0 : c0) * q123[v];  hoist the qubit-0 factor.
        float p23_0 = c2 * c3, p23_1 = c2 * s3, p23_2 = s2 * c3, p23_3 = s2 * s3;
        float dot;
        dot  = (c1 * p23_0) * acc[0];
        dot  = fmaf(c1 * p23_1, acc[1], dot);
        dot  = fmaf(c1 * p23_2, acc[2], dot);
        dot  = fmaf(c1 * p23_3, acc[3], dot);
        dot  = fmaf(s1 * p23_0, acc[4], dot);
        dot  = fmaf(s1 * p23_1, acc[5], dot);
        dot  = fmaf(s1 * p23_2, acc[6], dot);
        dot  = fmaf(s1 * p23_3, acc[7], dot);
        float partial = (h ? s0 : c0) * dot;

        // combine the two j-halves living in lanes l and l+16
        float full = partial + __shfl_xor(partial, 16, 32);

        if (h == 0 && (base + l) < B) out[base + l] = full;  // 64B coalesced
    }
}

extern "C" void kernel_launch(void* const* d_in, const int* in_sizes, int n_in,
                              void* d_out, int out_size, void* d_ws, size_t ws_size,
                              hipStream_t stream)
{
    const float* x = (const float*)d_in[0];   // [B,4]
    const float* w = (const float*)d_in[1];   // [8]
    float* out = (float*)d_out;               // [B]
    float* A   = (float*)d_ws;                // 16x16 f32 scratch

    int B = in_sizes[0] / 4;

    qc_setup<<<1, 256, 0, stream>>>(w, A);

    int tiles = (B + 15) / 16;
    int tilesPerBlock = WAVES_PER_BLOCK * TILES_PER_WAVE;
    int blocks = (tiles + tilesPerBlock - 1) / tilesPerBlock;
    qc_main<<<blocks, 256, 0, stream>>>(x, A, out, B);
}